// spike_74801150427208
// MI455X (gfx1250) — compile-verified
//
#include <hip/hip_runtime.h>

// ---------------------------------------------------------------------------
// spike recurrence: out[t] = s0 @ (x - I)^t, t = 0..2047, N = 2048.
// Log-doubling: rows [k,2k) = rows [0,k) @ P where P = A^k; then P <- P*P.
// All GEMMs in fp32 on V_WMMA_F32_16X16X4_F32 (CDNA5 fp32 tensor op).
// Block tile 128x128 (8 waves, each 32x64 = 8 accumulators), KC = 32.
// B staged pair-interleaved so each B fragment is one aligned ds_load_b64.
// ---------------------------------------------------------------------------

#define NN      2048
#define TILE_M  128
#define TILE_N  128
#define KC      32
#define A_PAD   36    // LDS row stride for A tile (alignment + bank spread)
#define B_PADN  132   // LDS column count for B pair tile

typedef float v2f __attribute__((ext_vector_type(2)));
typedef float v8f __attribute__((ext_vector_type(8)));

// A = x - I
__global__ __launch_bounds__(256) void build_A_kernel(const float* __restrict__ x,
                                                      float* __restrict__ A) {
    int idx = blockIdx.x * blockDim.x + threadIdx.x;   // 0 .. 2048*2048-1
    int row = idx >> 11;
    int col = idx & (NN - 1);
    float v = x[idx];
    if (row == col) v -= 1.0f;
    A[idx] = v;
}

// out row 0 = start[199]
__global__ __launch_bounds__(256) void row0_kernel(const float* __restrict__ start,
                                                   float* __restrict__ S) {
    int j = blockIdx.x * blockDim.x + threadIdx.x;
    S[j] = start[199 * NN + j];
}

// C[M x 2048] = Amat[M x 2048] @ Bmat[2048 x 2048], fp32 WMMA.
// Rows of Amat beyond M read as zero; rows of C beyond M not written.
__global__ __launch_bounds__(256)
void gemm_f32_wmma(const float* __restrict__ Am, const float* __restrict__ Bm,
                   float* __restrict__ Cm, int M) {
    __shared__ float As[TILE_M][A_PAD];          // 128 x 36 floats = 18432 B
    __shared__ float Bp[KC / 2][B_PADN][2];      // 16 x 132 x 2 floats = 16896 B

    const int tid  = threadIdx.x;
    const int lane = tid & 31;
    const int wave = tid >> 5;
    const int wm   = wave & 3;            // wave M index: 0..3  (32 rows each)
    const int wn   = wave >> 2;           // wave N index: 0..1  (64 cols each)
    const int hi   = lane >> 4;           // half-wave select
    const int ml   = lane & 15;

    const int blockN0 = blockIdx.x * TILE_N;
    const int blockM0 = blockIdx.y * TILE_M;

    v8f a00 = {}, a01 = {}, a02 = {}, a03 = {};   // rows wm*32 +  0..15
    v8f a10 = {}, a11 = {}, a12 = {}, a13 = {};   // rows wm*32 + 16..31

    // staging coordinates
    const int rowA = tid >> 3;            // 0..31 (then +32, +64, +96)
    const int colA = (tid & 7) << 2;      // 0,4,...,28
    const int kp0  = tid >> 5;            // 0..7 (then +8)
    const int cBb  = (tid & 31) << 2;     // 0,4,...,124

    for (int kb = 0; kb < NN; kb += KC) {
        __syncthreads();
        // ---- stage A tile: rows blockM0 .. +127, cols kb .. kb+31 ----
        {
            float4 z = {0.f, 0.f, 0.f, 0.f};
#pragma unroll
            for (int i = 0; i < 4; ++i) {
                int r = blockM0 + rowA + 32 * i;
                float4 a = (r < M) ? *(const float4*)(Am + (size_t)r * NN + kb + colA) : z;
                *(float4*)&As[rowA + 32 * i][colA] = a;
            }
        }
        // ---- stage B tile pair-interleaved: rows kb..kb+31 -> Bp[k/2][n][k&1] ----
        {
#pragma unroll
            for (int i = 0; i < 2; ++i) {
                int kp = kp0 + 8 * i;     // pair index 0..15
                const float* brow = Bm + (size_t)(kb + 2 * kp) * NN + blockN0 + cBb;
                float4 lo = *(const float4*)(brow);
                float4 hi4 = *(const float4*)(brow + NN);
                v2f p0 = {lo.x, hi4.x}, p1 = {lo.y, hi4.y};
                v2f p2 = {lo.z, hi4.z}, p3 = {lo.w, hi4.w};
                *(v2f*)&Bp[kp][cBb + 0][0] = p0;
                *(v2f*)&Bp[kp][cBb + 1][0] = p1;
                *(v2f*)&Bp[kp][cBb + 2][0] = p2;
                *(v2f*)&Bp[kp][cBb + 3][0] = p3;
            }
        }
        __syncthreads();

        // ---- compute: 8 k-steps of 4; 2 A-frags x 4 B-frags = 8 WMMAs each ----
        const int cb0 = wn * 64 + ml;
        const int ar0 = wm * 32 + ml;
#pragma unroll
        for (int kk = 0; kk < KC; kk += 4) {
            // fp32 A fragment: lane holds A[m][k0+2*hi], A[m][k0+2*hi+1]
            v2f fa0 = *(const v2f*)&As[ar0][kk + 2 * hi];
            v2f fa1 = *(const v2f*)&As[ar0 + 16][kk + 2 * hi];
            // fp32 B fragment: rows (kk+2*hi, kk+2*hi+1) adjacent in pair layout
            int kB2 = (kk >> 1) + hi;
            v2f fb0 = *(const v2f*)&Bp[kB2][cb0 +  0][0];
            v2f fb1 = *(const v2f*)&Bp[kB2][cb0 + 16][0];
            v2f fb2 = *(const v2f*)&Bp[kB2][cb0 + 32][0];
            v2f fb3 = *(const v2f*)&Bp[kB2][cb0 + 48][0];
            a00 = __builtin_amdgcn_wmma_f32_16x16x4_f32(false, fa0, false, fb0, (short)0, a00, false, false);
            a01 = __builtin_amdgcn_wmma_f32_16x16x4_f32(false, fa0, false, fb1, (short)0, a01, false, false);
            a02 = __builtin_amdgcn_wmma_f32_16x16x4_f32(false, fa0, false, fb2, (short)0, a02, false, false);
            a03 = __builtin_amdgcn_wmma_f32_16x16x4_f32(false, fa0, false, fb3, (short)0, a03, false, false);
            a10 = __builtin_amdgcn_wmma_f32_16x16x4_f32(false, fa1, false, fb0, (short)0, a10, false, false);
            a11 = __builtin_amdgcn_wmma_f32_16x16x4_f32(false, fa1, false, fb1, (short)0, a11, false, false);
            a12 = __builtin_amdgcn_wmma_f32_16x16x4_f32(false, fa1, false, fb2, (short)0, a12, false, false);
            a13 = __builtin_amdgcn_wmma_f32_16x16x4_f32(false, fa1, false, fb3, (short)0, a13, false, false);
        }
    }

    // ---- epilogue: C layout = VGPR r -> row r (+8 for hi lanes), col = lane%16 ----
    const int colBase = blockN0 + wn * 64 + ml;
    const int row0 = blockM0 + wm * 32 + 8 * hi;
#pragma unroll
    for (int r = 0; r < 8; ++r) {
        int rg = row0 + r;
        if (rg < M) {
            float* cp = Cm + (size_t)rg * NN + colBase;
            cp[ 0] = a00[r];
            cp[16] = a01[r];
            cp[32] = a02[r];
            cp[48] = a03[r];
        }
        int rg2 = row0 + 16 + r;
        if (rg2 < M) {
            float* cp = Cm + (size_t)rg2 * NN + colBase;
            cp[ 0] = a10[r];
            cp[16] = a11[r];
            cp[32] = a12[r];
            cp[48] = a13[r];
        }
    }
}

extern "C" void kernel_launch(void* const* d_in, const int* in_sizes, int n_in,
                              void* d_out, int out_size, void* d_ws, size_t ws_size,
                              hipStream_t stream) {
    (void)in_sizes; (void)n_in; (void)out_size;
    const float* x     = (const float*)d_in[0];   // (2048, 2048) fp32
    const float* start = (const float*)d_in[1];   // (200, 2048) fp32
    float* S = (float*)d_out;                     // (2048, 2048) fp32

    const size_t MATSZ = (size_t)NN * NN;         // elements
    if (ws_size < 2 * MATSZ * sizeof(float)) return;  // need 32 MB scratch
    float* P  = (float*)d_ws;                     // current power A^k
    float* Pn = P + MATSZ;                        // next power buffer

    // P = x - I ; S[0] = start[199]
    build_A_kernel<<<dim3((unsigned)(MATSZ / 256)), dim3(256), 0, stream>>>(x, P);
    row0_kernel<<<dim3(NN / 256), dim3(256), 0, stream>>>(start, S);

    // log-doubling: 11 row-block GEMMs + 10 squarings, all stream-ordered
    for (int k = 1; k < NN; k <<= 1) {
        dim3 grid(NN / TILE_N, (unsigned)((k + TILE_M - 1) / TILE_M));
        gemm_f32_wmma<<<grid, dim3(256), 0, stream>>>(S, P, S + (size_t)k * NN, k);
        if (2 * k < NN) {
            dim3 g2(NN / TILE_N, NN / TILE_M);
            gemm_f32_wmma<<<g2, dim3(256), 0, stream>>>(P, P, Pn, NN);
            float* t = P; P = Pn; Pn = t;
        }
    }
}